// DCG_19413252178428
// MI455X (gfx1250) — compile-verified
//
#include <hip/hip_runtime.h>
#include <math.h>

// Problem constants (from the reference)
#define NN   2000
#define AA   16
#define OBSD 64
#define HD   128
#define ED   16000
#define MPIT 8
#define CHUNK 8000          // p-MLP row chunk (4 chunks cover 2E = 32000 rows)

typedef __attribute__((ext_vector_type(2))) float v2f;
typedef __attribute__((ext_vector_type(8))) float v8f;

// ---------------------------------------------------------------------------
// Register-blocked fp32 WMMA GEMM:
//   Y[M,Nout] = relu?(X[M,K] @ W[Nout,K]^T + bias) * scale
// One wave computes an MB x NB grid of 16x16 C tiles with
// V_WMMA_F32_16X16X4_F32, K stepped by 4.  A fragment: lane l<16 holds row
// m0+l, K=k0+{0,1}; lanes 16-31 hold K=k0+{2,3}.  B mirrored over N.
// Blocking raises WMMA-per-load: MB*NB wmma for MB+NB b64 loads per K-step.
// ---------------------------------------------------------------------------
template<int MB, int NB>
__global__ void k_gemm_t(const float* __restrict__ X, const float* __restrict__ W,
                         const float* __restrict__ bias, float* __restrict__ Y,
                         int K, int Nout, float scale, int relu) {
  int m0   = blockIdx.x * (16 * MB);
  int n0   = blockIdx.y * (16 * NB);
  int lane = threadIdx.x;
  int half = lane >> 4;
  int l    = lane & 15;
  const float* xrow[MB];
  const float* wrow[NB];
#pragma unroll
  for (int mb = 0; mb < MB; ++mb) xrow[mb] = X + (size_t)(m0 + mb * 16 + l) * K;
#pragma unroll
  for (int nb = 0; nb < NB; ++nb) wrow[nb] = W + (size_t)(n0 + nb * 16 + l) * K;

  const v8f vzero = {0.f,0.f,0.f,0.f,0.f,0.f,0.f,0.f};
  v8f acc[MB][NB];
#pragma unroll
  for (int mb = 0; mb < MB; ++mb)
#pragma unroll
    for (int nb = 0; nb < NB; ++nb) acc[mb][nb] = vzero;

  for (int k0 = 0; k0 < K; k0 += 4) {
    int kk = k0 + 2 * half;
    v2f a[MB], b[NB];
#pragma unroll
    for (int mb = 0; mb < MB; ++mb) { a[mb].x = xrow[mb][kk]; a[mb].y = xrow[mb][kk + 1]; }
#pragma unroll
    for (int nb = 0; nb < NB; ++nb) { b[nb].x = wrow[nb][kk]; b[nb].y = wrow[nb][kk + 1]; }
#pragma unroll
    for (int mb = 0; mb < MB; ++mb)
#pragma unroll
      for (int nb = 0; nb < NB; ++nb)
        acc[mb][nb] = __builtin_amdgcn_wmma_f32_16x16x4_f32(false, a[mb], false, b[nb],
                                                            (short)0, acc[mb][nb],
                                                            false, false);
  }

  float bb[NB];
#pragma unroll
  for (int nb = 0; nb < NB; ++nb) bb[nb] = bias[n0 + nb * 16 + l];
#pragma unroll
  for (int mb = 0; mb < MB; ++mb)
#pragma unroll
    for (int nb = 0; nb < NB; ++nb) {
      int n = n0 + nb * 16 + l;
#pragma unroll
      for (int r = 0; r < 8; ++r) {
        int m = m0 + mb * 16 + r + half * 8;   // C/D layout: vgpr r -> M=r (+8 hi half)
        float v = acc[mb][nb][r] + bb[nb];
        if (relu) v = fmaxf(v, 0.0f);
        Y[(size_t)m * Nout + n] = v * scale;
      }
    }
}

// ---------------------------------------------------------------------------
// p-MLP layer0: virtual input row r (global, in [0,2E)) = concat(h[i], h[j])
//   r <  E : i = edges[r].x, j = edges[r].y      (s_ij)
//   r >= E : i = edges[r-E].y, j = edges[r-E].x  (s_ji)
// K = 256 (two halves of 128), ReLU.  MB=2 x NB=4 register blocking.
// ---------------------------------------------------------------------------
#define CMB 2
#define CNB 4
__global__ void k_gemm_concat(const float* __restrict__ h, const int* __restrict__ edges,
                              const float* __restrict__ W, const float* __restrict__ bias,
                              float* __restrict__ Y, int rowBase, int Nout) {
  int m0   = blockIdx.x * (16 * CMB);
  int n0   = blockIdx.y * (16 * CNB);
  int lane = threadIdx.x;
  int half = lane >> 4;
  int l    = lane & 15;

  const float* lo[CMB];
  const float* hi[CMB];
#pragma unroll
  for (int mb = 0; mb < CMB; ++mb) {
    int r = rowBase + m0 + mb * 16 + l;        // A-fragment row for this lane
    int e, i, j;
    if (r < ED) { e = r;      i = edges[2*e];   j = edges[2*e+1]; }
    else        { e = r - ED; i = edges[2*e+1]; j = edges[2*e];   }
    lo[mb] = h + (size_t)i * HD;
    hi[mb] = h + (size_t)j * HD;
  }
  const float* wrow[CNB];
#pragma unroll
  for (int nb = 0; nb < CNB; ++nb) wrow[nb] = W + (size_t)(n0 + nb * 16 + l) * 256;

  const v8f vzero = {0.f,0.f,0.f,0.f,0.f,0.f,0.f,0.f};
  v8f acc[CMB][CNB];
#pragma unroll
  for (int mb = 0; mb < CMB; ++mb)
#pragma unroll
    for (int nb = 0; nb < CNB; ++nb) acc[mb][nb] = vzero;

  for (int k0 = 0; k0 < 256; k0 += 4) {
    int kk = k0 + 2 * half;                    // even; pair never straddles 128 boundary
    v2f a[CMB], b[CNB];
#pragma unroll
    for (int mb = 0; mb < CMB; ++mb) {
      const float* src = (kk < 128) ? (lo[mb] + kk) : (hi[mb] + kk - 128);
      a[mb].x = src[0]; a[mb].y = src[1];
    }
#pragma unroll
    for (int nb = 0; nb < CNB; ++nb) { b[nb].x = wrow[nb][kk]; b[nb].y = wrow[nb][kk + 1]; }
#pragma unroll
    for (int mb = 0; mb < CMB; ++mb)
#pragma unroll
      for (int nb = 0; nb < CNB; ++nb)
        acc[mb][nb] = __builtin_amdgcn_wmma_f32_16x16x4_f32(false, a[mb], false, b[nb],
                                                            (short)0, acc[mb][nb],
                                                            false, false);
  }

  float bb[CNB];
#pragma unroll
  for (int nb = 0; nb < CNB; ++nb) bb[nb] = bias[n0 + nb * 16 + l];
#pragma unroll
  for (int mb = 0; mb < CMB; ++mb)
#pragma unroll
    for (int nb = 0; nb < CNB; ++nb) {
      int n = n0 + nb * 16 + l;
#pragma unroll
      for (int r = 0; r < 8; ++r) {
        int m = m0 + mb * 16 + r + half * 8;
        float v = fmaxf(acc[mb][nb][r] + bb[nb], 0.0f);
        Y[(size_t)m * Nout + n] = v;
      }
    }
}

// p output accumulate: p[e] = 0.5*(raw_ij[e] + raw_ji[e])   (raw already /E)
__global__ void k_p_accum(const float* __restrict__ raw, float* __restrict__ pout,
                          int rowBase) {
  int idx = blockIdx.x * blockDim.x + threadIdx.x;
  if (idx >= CHUNK * 256) return;
  int lr = idx >> 8, t = idx & 255;
  int r = rowBase + lr;
  float v = 0.5f * raw[(size_t)lr * 256 + t];
  if (r < ED) pout[(size_t)r * 256 + t] = v;                     // first direction
  else        pout[(size_t)(r - ED) * 256 + t] += v;             // second direction
}

// enc = concat(x, prev_action)  [N, 80]
__global__ void k_enc(const float* __restrict__ x, const float* __restrict__ pa,
                      float* __restrict__ enc) {
  int idx = blockIdx.x * blockDim.x + threadIdx.x;
  if (idx >= NN * 80) return;
  int n = idx / 80, k = idx % 80;
  enc[idx] = (k < OBSD) ? x[n * OBSD + k] : pa[n * AA + (k - OBSD)];
}

// GRU gate combine: h = (1-z)*tanh(gin + r*ghn) + z*h_prev
__global__ void k_gru(const float* __restrict__ gi, const float* __restrict__ gh,
                      const float* __restrict__ hprev, float* __restrict__ h,
                      float* __restrict__ stateOut) {
  int idx = blockIdx.x * blockDim.x + threadIdx.x;
  if (idx >= NN * HD) return;
  int n = idx / HD, c = idx % HD;
  const float* gin_ = gi + (size_t)n * 384;
  const float* ghn_ = gh + (size_t)n * 384;
  float r  = 1.0f / (1.0f + expf(-(gin_[c] + ghn_[c])));
  float z  = 1.0f / (1.0f + expf(-(gin_[HD + c] + ghn_[HD + c])));
  float nn = tanhf(gin_[2*HD + c] + r * ghn_[2*HD + c]);
  float hv = (1.0f - z) * nn + z * hprev[idx];
  h[idx] = hv;
  stateOut[idx] = hv;
}

__global__ void k_fill_f(float* p, float v, int n) {
  int i = blockIdx.x * blockDim.x + threadIdx.x;
  if (i < n) p[i] = v;
}
__global__ void k_fill_i(int* p, int v, int n) {
  int i = blockIdx.x * blockDim.x + threadIdx.x;
  if (i < n) p[i] = v;
}

// ---- Edge dedup (reference uses a dense [N,N,A] M: duplicate (i,j) edges
// ---- collapse to one cell; S must count each cell once) ----
__global__ void k_owner(const int* __restrict__ edges, int* owner) {
  int e = blockIdx.x * blockDim.x + threadIdx.x;
  if (e >= ED) return;
  atomicMin(&owner[edges[2*e] * NN + edges[2*e+1]], e);
}
__global__ void k_primary(const int* __restrict__ edges, const int* __restrict__ owner,
                          int* __restrict__ primary) {
  int e = blockIdx.x * blockDim.x + threadIdx.x;
  if (e >= ED) return;
  primary[e] = (owner[edges[2*e] * NN + edges[2*e+1]] == e) ? 1 : 0;
}
__global__ void k_deg(const int* __restrict__ edges, const int* __restrict__ primary,
                      int* deg) {
  int e = blockIdx.x * blockDim.x + threadIdx.x;
  if (e >= ED || !primary[e]) return;
  atomicAdd(&deg[edges[2*e]], 1);
  atomicAdd(&deg[edges[2*e+1]], 1);
}
__global__ void k_prefix(const int* __restrict__ deg, int* rowstart) {
  if (blockIdx.x || threadIdx.x) return;
  int s = 0;
  for (int n = 0; n < NN; ++n) { rowstart[n] = s; s += deg[n]; }
  rowstart[NN] = s;
}
// Deterministic CSR fill: slot = count of earlier primary edges at same node.
__global__ void k_rankfill(const int* __restrict__ edges, const int* __restrict__ primary,
                           const int* __restrict__ rowstart, int* __restrict__ adj) {
  int e = blockIdx.x * blockDim.x + threadIdx.x;
  if (e >= ED || !primary[e]) return;
  int i = edges[2*e], j = edges[2*e+1];
  int ri = 0, rj = 0;
  for (int ep = 0; ep < e; ++ep) {
    if (!primary[ep]) continue;
    int a = edges[2*ep], b = edges[2*ep+1];
    ri += (a == i) || (b == i);
    rj += (a == j) || (b == j);
  }
  adj[rowstart[i] + ri] = e * 2 + 0;   // side 0: message toward i  (msgI)
  adj[rowstart[j] + rj] = e * 2 + 1;   // side 1: message toward j  (msgJ)
}

// ---- Max-sum message passing ----
// 16 lanes per edge; lane t = action index.
// msg_to_j[t] = max_ai(p[ai][t] + base_i[ai]) - mean_t
// msg_to_i[t] = max_ai(p[ai][t]) + base_j[t]  - mean_t   (exactly per reference)
__global__ void k_edge_update(const float* __restrict__ u, const float* __restrict__ p,
                              const int* __restrict__ edges,
                              float* __restrict__ msgI, float* __restrict__ msgJ,
                              const float* __restrict__ S) {
  int t = threadIdx.x & 15;
  int e = blockIdx.x * 16 + (threadIdx.x >> 4);
  int i = edges[2*e], j = edges[2*e+1];
  float mi = msgI[e * AA + t], mj = msgJ[e * AA + t];
  float base_i = u[i * AA + t] + S[i * AA + t] - mi;
  float base_j = u[j * AA + t] + S[j * AA + t] - mj;
  const float* pe = p + (size_t)e * 256;
  float m1 = -3.402823466e38f, cm = -3.402823466e38f;
  for (int ai = 0; ai < 16; ++ai) {
    float bi = __shfl(base_i, ai, 16);
    float pv = pe[ai * 16 + t];
    m1 = fmaxf(m1, pv + bi);
    cm = fmaxf(cm, pv);
  }
  float m2 = cm + base_j;
  float s1 = m1, s2 = m2;
  for (int o = 8; o; o >>= 1) {
    s1 += __shfl_xor(s1, o, 16);
    s2 += __shfl_xor(s2, o, 16);
  }
  msgJ[e * AA + t] = m1 - s1 * (1.0f / 16.0f);
  msgI[e * AA + t] = m2 - s2 * (1.0f / 16.0f);
}

// S[n,a] = sum of incoming messages (deterministic CSR gather, dedup'd)
__global__ void k_S(const int* __restrict__ rowstart, const int* __restrict__ adj,
                    const float* __restrict__ msgI, const float* __restrict__ msgJ,
                    float* __restrict__ S) {
  int idx = blockIdx.x * blockDim.x + threadIdx.x;
  if (idx >= NN * AA) return;
  int n = idx >> 4, a = idx & 15;
  float acc = 0.0f;
  int kend = rowstart[n + 1];
  for (int k = rowstart[n]; k < kend; ++k) {
    int v = adj[k]; int e = v >> 1;
    acc += (v & 1) ? msgJ[e * AA + a] : msgI[e * AA + a];
  }
  S[idx] = acc;
}

__global__ void k_argmax(const float* __restrict__ u, const float* __restrict__ S,
                         int* __restrict__ aidx) {
  int n = blockIdx.x * blockDim.x + threadIdx.x;
  if (n >= NN) return;
  float best = u[n * AA] + S[n * AA];
  int bi = 0;
  for (int a = 1; a < AA; ++a) {
    float v = u[n * AA + a] + S[n * AA + a];
    if (v > best) { best = v; bi = a; }   // first max wins (jnp.argmax semantics)
  }
  aidx[n] = bi;
}

// q_real partial sums over N node terms + E edge terms
__global__ void k_qpart(const float* __restrict__ u, const float* __restrict__ p,
                        const int* __restrict__ edges, const int* __restrict__ aidx,
                        float* __restrict__ partial) {
  __shared__ float sd[256];
  int idx = blockIdx.x * 256 + threadIdx.x;
  float v = 0.0f;
  if (idx < NN) v = u[idx * AA + aidx[idx]];
  else if (idx < NN + ED) {
    int e = idx - NN;
    int i = edges[2*e], j = edges[2*e+1];
    v = p[(size_t)e * 256 + aidx[i] * 16 + aidx[j]];
  }
  sd[threadIdx.x] = v;
  __syncthreads();
  for (int o = 128; o; o >>= 1) {
    if (threadIdx.x < o) sd[threadIdx.x] += sd[threadIdx.x + o];
    __syncthreads();
  }
  if (threadIdx.x == 0) partial[blockIdx.x] = sd[0];
}

__global__ void k_qfinal(const float* __restrict__ partial, int nPart,
                         float* qmax, int* flag, float* qout) {
  if (blockIdx.x || threadIdx.x) return;
  float q = 0.0f;
  for (int b = 0; b < nPart; ++b) q += partial[b];
  int better = q > qmax[0];
  if (better) qmax[0] = q;
  flag[0] = better;
  qout[0] = qmax[0];
}

__global__ void k_opta(const int* __restrict__ flag, const int* __restrict__ aidx,
                       float* __restrict__ aout) {
  if (!flag[0]) return;
  int idx = blockIdx.x * blockDim.x + threadIdx.x;
  if (idx >= NN * AA) return;
  aout[idx] = ((idx & 15) == aidx[idx >> 4]) ? 1.0f : 0.0f;
}

// ---------------------------------------------------------------------------
extern "C" void kernel_launch(void* const* d_in, const int* in_sizes, int n_in,
                              void* d_out, int out_size, void* d_ws, size_t ws_size,
                              hipStream_t stream) {
  (void)in_sizes; (void)n_in; (void)out_size; (void)ws_size;
  const float* x    = (const float*)d_in[0];
  const float* pa   = (const float*)d_in[1];
  const float* st   = (const float*)d_in[2];
  const int*   edges= (const int*)d_in[3];
  const float* wih  = (const float*)d_in[4];
  const float* whh  = (const float*)d_in[5];
  const float* bih  = (const float*)d_in[6];
  const float* bhh  = (const float*)d_in[7];
  const float* uW0  = (const float*)d_in[8];  const float* ub0 = (const float*)d_in[9];
  const float* uW1  = (const float*)d_in[10]; const float* ub1 = (const float*)d_in[11];
  const float* uW2  = (const float*)d_in[12]; const float* ub2 = (const float*)d_in[13];
  const float* pW0  = (const float*)d_in[14]; const float* pb0 = (const float*)d_in[15];
  const float* pW1  = (const float*)d_in[16]; const float* pb1 = (const float*)d_in[17];
  const float* pW2  = (const float*)d_in[18]; const float* pb2 = (const float*)d_in[19];
  const float* pW3  = (const float*)d_in[20]; const float* pb3 = (const float*)d_in[21];
  const float* pW4  = (const float*)d_in[22]; const float* pb4 = (const float*)d_in[23];

  // Output layout (flat fp32, reference return order)
  float* out      = (float*)d_out;
  float* outA     = out;            // [N,A]        32000
  float* outQ     = out + 32000;    // scalar           1
  float* outU     = out + 32001;    // [1,N,A]      32000
  float* outP     = out + 64001;    // [E,A,A]    4096000
  float* outState = out + 4160001;  // [1,1,N,H]   256000

  // Workspace layout (~47 MB)
  float* wf   = (float*)d_ws;
  float* enc  = wf;                 // 160000
  float* gi   = wf + 160000;        // 768000
  float* gh   = wf + 928000;        // 768000
  float* h    = wf + 1696000;       // 256000
  float* ua1  = wf + 1952000;       // 512000
  float* ua2  = wf + 2464000;       // 512000
  float* pBufA= wf + 2976000;       // 2048000 (CHUNK x 256)
  float* pBufB= wf + 5024000;       // 2048000
  float* msgI = wf + 7072000;       // 256000
  float* msgJ = wf + 7328000;       // 256000
  float* Sb   = wf + 7584000;       // 32000
  float* part = wf + 7616000;       // 128
  float* qmax = wf + 7616128;       // 4 (pad)
  int*   wi      = (int*)(wf + 7616132);
  int*   owner   = wi;              // 4,000,000
  int*   deg     = wi + 4000000;    // 2000
  int*   rowstart= wi + 4002000;    // 2001 (pad to 2004)
  int*   adj     = wi + 4004004;    // <= 32000
  int*   aidx    = wi + 4036004;    // 2000
  int*   primary = wi + 4038004;    // 16000
  int*   flag    = wi + 4054004;    // 1

  dim3 wave(32);

  // ---- init outputs that may persist zeros ----
  k_fill_f<<<125, 256, 0, stream>>>(outA, 0.0f, NN * AA);
  k_fill_f<<<1, 1, 0, stream>>>(qmax, 0.0f, 1);

  // ---- GRU encoder ----
  k_enc<<<625, 256, 0, stream>>>(x, pa, enc);
  k_gemm_t<1,4><<<dim3(125, 6), wave, 0, stream>>>(enc, wih, bih, gi, 80, 384, 1.0f, 0);
  k_gemm_t<1,4><<<dim3(125, 6), wave, 0, stream>>>(st,  whh, bhh, gh, HD, 384, 1.0f, 0);
  k_gru<<<1000, 256, 0, stream>>>(gi, gh, st, h, outState);

  // ---- u MLP: 128 -> 256 -> 256 -> 16, /N ----
  k_gemm_t<1,4><<<dim3(125, 4), wave, 0, stream>>>(h,   uW0, ub0, ua1, HD,  256, 1.0f, 1);
  k_gemm_t<1,4><<<dim3(125, 4), wave, 0, stream>>>(ua1, uW1, ub1, ua2, 256, 256, 1.0f, 1);
  k_gemm_t<1,1><<<dim3(125, 1), wave, 0, stream>>>(ua2, uW2, ub2, outU, 256, AA,
                                                   1.0f / (float)NN, 0);

  // ---- p MLP (both edge directions, chunked): 256->128->256->256->16->256, /E ----
  for (int c = 0; c < 4; ++c) {
    int rb = c * CHUNK;
    k_gemm_concat<<<dim3(CHUNK/32, 2), wave, 0, stream>>>(h, edges, pW0, pb0, pBufA, rb, 128);
    k_gemm_t<2,4><<<dim3(CHUNK/32, 4), wave, 0, stream>>>(pBufA, pW1, pb1, pBufB, 128, 256, 1.0f, 1);
    k_gemm_t<2,4><<<dim3(CHUNK/32, 4), wave, 0, stream>>>(pBufB, pW2, pb2, pBufA, 256, 256, 1.0f, 1);
    k_gemm_t<2,1><<<dim3(CHUNK/32, 1), wave, 0, stream>>>(pBufA, pW3, pb3, pBufB, 256, AA, 1.0f, 1);
    k_gemm_t<2,4><<<dim3(CHUNK/32, 4), wave, 0, stream>>>(pBufB, pW4, pb4, pBufA, AA, 256,
                                                          1.0f / (float)ED, 0);
    k_p_accum<<<CHUNK, 256, 0, stream>>>(pBufA, outP, rb);
  }

  // ---- edge dedup + deterministic CSR adjacency (one-time) ----
  k_fill_i<<<15625, 256, 0, stream>>>(owner, 0x7fffffff, NN * NN);
  k_owner<<<63, 256, 0, stream>>>(edges, owner);
  k_primary<<<63, 256, 0, stream>>>(edges, owner, primary);
  k_fill_i<<<8, 256, 0, stream>>>(deg, 0, NN);
  k_deg<<<63, 256, 0, stream>>>(edges, primary, deg);
  k_prefix<<<1, 1, 0, stream>>>(deg, rowstart);
  k_rankfill<<<63, 256, 0, stream>>>(edges, primary, rowstart, adj);

  // ---- message passing init ----
  k_fill_f<<<1000, 256, 0, stream>>>(msgI, 0.0f, ED * AA);
  k_fill_f<<<1000, 256, 0, stream>>>(msgJ, 0.0f, ED * AA);
  k_fill_f<<<125, 256, 0, stream>>>(Sb, 0.0f, NN * AA);

  // ---- 8 max-sum iterations ----
  for (int it = 0; it < MPIT; ++it) {
    k_edge_update<<<ED/16, 256, 0, stream>>>(outU, outP, edges, msgI, msgJ, Sb);
    k_S<<<125, 256, 0, stream>>>(rowstart, adj, msgI, msgJ, Sb);
    k_argmax<<<8, 256, 0, stream>>>(outU, Sb, aidx);
    k_qpart<<<71, 256, 0, stream>>>(outU, outP, edges, aidx, part);
    k_qfinal<<<1, 1, 0, stream>>>(part, 71, qmax, flag, outQ);
    k_opta<<<125, 256, 0, stream>>>(flag, aidx, outA);
  }
}